// LSTM_52140902973763
// MI455X (gfx1250) — compile-verified
//
#include <hip/hip_runtime.h>
#include <math.h>

// Problem constants (from reference)
#define SEQ   256
#define BATCH 32
#define NIN   1024
#define NH    1024
#define NOUT  32000
#define GDIM  (4 * NH)        // 4096 gate columns
#define MROWS (SEQ * BATCH)   // 8192

typedef float v2f __attribute__((ext_vector_type(2)));
typedef float v8f __attribute__((ext_vector_type(8)));

// ---------------------------------------------------------------------------
// CDNA5 f32 WMMA: D(16x16) = A(16x4) * B(4x16) + C
// A/B: 2 VGPRs per lane; element (m,k): lane = m + 16*(k>>1), vgpr = k&1
// C/D: 8 VGPRs; element (m,n): n = lane%16, m = r + 8*(lane>>4)
// ---------------------------------------------------------------------------
__device__ __forceinline__ v8f wmma_f32_k4(v2f a, v2f b, v8f c) {
    return __builtin_amdgcn_wmma_f32_16x16x4_f32(
        /*neg_a=*/false, a, /*neg_b=*/false, b,
        /*c_mod=*/(short)0, c, /*reuse_a=*/false, /*reuse_b=*/false);
}

__device__ __forceinline__ float sig_(float x) { return 1.0f / (1.0f + __expf(-x)); }

// ---------------------------------------------------------------------------
// Generic GEMM: Out[M x N] = A[M x K] @ W[N x K]^T  (+ bias1[n] + bias2[n])
// Block: 256 threads = 8 waves; each wave owns 2 M-tiles x 4 N-tiles.
// Pipelined K loop with peeled epilogue (no per-iteration select).
// grid.x = N/64, grid.y = M/256
// ---------------------------------------------------------------------------
__global__ __launch_bounds__(256)
void gemm_nt_wmma(const float* __restrict__ A, const float* __restrict__ W,
                  const float* __restrict__ bias1, const float* __restrict__ bias2,
                  float* __restrict__ Out, int M, int N, int K) {
    const int lane   = threadIdx.x & 31;
    const int wave   = threadIdx.x >> 5;
    const int m_base = (blockIdx.y * 16 + wave * 2) * 16;   // 32 rows per wave
    const int n_base = blockIdx.x * 64;
    if (m_base >= M) return;
    const int lm   = lane & 15;
    const int koff = (lane >> 4) * 2;

    const float* a0p = A + (size_t)(m_base +  0 + lm) * K + koff;
    const float* a1p = A + (size_t)(m_base + 16 + lm) * K + koff;
    const float* w0p = W + (size_t)(n_base +  0 + lm) * K + koff;
    const float* w1p = W + (size_t)(n_base + 16 + lm) * K + koff;
    const float* w2p = W + (size_t)(n_base + 32 + lm) * K + koff;
    const float* w3p = W + (size_t)(n_base + 48 + lm) * K + koff;

    v8f acc00 = {}, acc01 = {}, acc02 = {}, acc03 = {};
    v8f acc10 = {}, acc11 = {}, acc12 = {}, acc13 = {};

    // Pipeline prologue: fragments for k = 0
    v2f a0c = *(const v2f*)(a0p);
    v2f a1c = *(const v2f*)(a1p);
    v2f b0c = *(const v2f*)(w0p);
    v2f b1c = *(const v2f*)(w1p);
    v2f b2c = *(const v2f*)(w2p);
    v2f b3c = *(const v2f*)(w3p);

    for (int k0 = 0; k0 < K - 4; k0 += 4) {
        v2f a0n = *(const v2f*)(a0p + k0 + 4);
        v2f a1n = *(const v2f*)(a1p + k0 + 4);
        v2f b0n = *(const v2f*)(w0p + k0 + 4);
        v2f b1n = *(const v2f*)(w1p + k0 + 4);
        v2f b2n = *(const v2f*)(w2p + k0 + 4);
        v2f b3n = *(const v2f*)(w3p + k0 + 4);

        acc00 = wmma_f32_k4(a0c, b0c, acc00);
        acc01 = wmma_f32_k4(a0c, b1c, acc01);
        acc02 = wmma_f32_k4(a0c, b2c, acc02);
        acc03 = wmma_f32_k4(a0c, b3c, acc03);
        acc10 = wmma_f32_k4(a1c, b0c, acc10);
        acc11 = wmma_f32_k4(a1c, b1c, acc11);
        acc12 = wmma_f32_k4(a1c, b2c, acc12);
        acc13 = wmma_f32_k4(a1c, b3c, acc13);

        a0c = a0n; a1c = a1n;
        b0c = b0n; b1c = b1n; b2c = b2n; b3c = b3n;
    }
    // Epilogue: last k-step
    acc00 = wmma_f32_k4(a0c, b0c, acc00);
    acc01 = wmma_f32_k4(a0c, b1c, acc01);
    acc02 = wmma_f32_k4(a0c, b2c, acc02);
    acc03 = wmma_f32_k4(a0c, b3c, acc03);
    acc10 = wmma_f32_k4(a1c, b0c, acc10);
    acc11 = wmma_f32_k4(a1c, b1c, acc11);
    acc12 = wmma_f32_k4(a1c, b2c, acc12);
    acc13 = wmma_f32_k4(a1c, b3c, acc13);

    float bv[4];
#pragma unroll
    for (int j = 0; j < 4; ++j) {
        int col = n_base + j * 16 + lm;
        float b = 0.0f;
        if (bias1) b += bias1[col];
        if (bias2) b += bias2[col];
        bv[j] = b;
    }
    const int row_half = (lane >> 4) * 8;
#pragma unroll
    for (int r = 0; r < 8; ++r) {
        float* o0 = Out + (size_t)(m_base + r + row_half) * N + n_base;
        float* o1 = Out + (size_t)(m_base + 16 + r + row_half) * N + n_base;
        o0[ 0 + lm] = acc00[r] + bv[0];
        o0[16 + lm] = acc01[r] + bv[1];
        o0[32 + lm] = acc02[r] + bv[2];
        o0[48 + lm] = acc03[r] + bv[3];
        o1[ 0 + lm] = acc10[r] + bv[0];
        o1[16 + lm] = acc11[r] + bv[1];
        o1[32 + lm] = acc12[r] + bv[2];
        o1[48 + lm] = acc13[r] + bv[3];
    }
}

// ---------------------------------------------------------------------------
// LSTM recurrence steps. 64 workgroups (one per 16 hidden units), 512 threads
// = 16 waves: wave w -> gate = w&3, K-quarter = w>>2. Each wave carries both
// batch halves (2 acc chains). 4-way K split quarters the serial dependent
// chain and doubles resident waves per SIMD for latency hiding. Partial sums
// are combined in LDS by the elementwise pass.
// ---------------------------------------------------------------------------
#define KQ (NH / 4)   // 256

__device__ __forceinline__ void wmma_pair_loop(
        const float* __restrict__ Arows, const float* __restrict__ Wrow,
        int kbeg, int kend, v8f& acc0, v8f& acc1, int lm, int koff) {
    const float* a0p = Arows + (size_t)( 0 + lm) * NH + koff;
    const float* a1p = Arows + (size_t)(16 + lm) * NH + koff;
    const float* wp  = Wrow + koff;
    v2f a0c = *(const v2f*)(a0p + kbeg);
    v2f a1c = *(const v2f*)(a1p + kbeg);
    v2f bc  = *(const v2f*)(wp  + kbeg);
    for (int k0 = kbeg; k0 < kend - 4; k0 += 4) {
        v2f a0n = *(const v2f*)(a0p + k0 + 4);
        v2f a1n = *(const v2f*)(a1p + k0 + 4);
        v2f bn  = *(const v2f*)(wp  + k0 + 4);
        acc0 = wmma_f32_k4(a0c, bc, acc0);
        acc1 = wmma_f32_k4(a1c, bc, acc1);
        a0c = a0n; a1c = a1n; bc = bn;
    }
    acc0 = wmma_f32_k4(a0c, bc, acc0);
    acc1 = wmma_f32_k4(a1c, bc, acc1);
}

__global__ __launch_bounds__(512)
void lstm_step_l0(const float* __restrict__ Xpre_t,  // [B x GDIM] (biases folded)
                  const float* __restrict__ Whh,     // [GDIM x NH]
                  const float* __restrict__ h_prev, const float* __restrict__ c_prev,
                  float* __restrict__ h_next, float* __restrict__ c_next) {
    __shared__ float g_lds[4][4][BATCH][16];   // [kq][gate][b][n] = 32 KB
    const int lane = threadIdx.x & 31;
    const int wave = threadIdx.x >> 5;
    const int gate = wave & 3;
    const int kq   = wave >> 2;
    const int n_base = blockIdx.x * 16;
    const int col_base = gate * NH + n_base;
    const int lm   = lane & 15;
    const int koff = (lane >> 4) * 2;

    v8f acc0 = {}, acc1 = {};
    wmma_pair_loop(h_prev, Whh + (size_t)(col_base + lm) * NH,
                   kq * KQ, kq * KQ + KQ, acc0, acc1, lm, koff);

    const int row_half = (lane >> 4) * 8;
#pragma unroll
    for (int r = 0; r < 8; ++r) {
        g_lds[kq][gate][r + row_half][lm]      = acc0[r];
        g_lds[kq][gate][16 + r + row_half][lm] = acc1[r];
    }
    __syncthreads();
    {
        int e = threadIdx.x;            // exactly BATCH*16 = 512 elements
        int m = e >> 4, n = e & 15;
        const float* xp = Xpre_t + (size_t)m * GDIM + n_base + n;
        float gi = g_lds[0][0][m][n] + g_lds[1][0][m][n] + g_lds[2][0][m][n] + g_lds[3][0][m][n] + xp[0 * NH];
        float gf = g_lds[0][1][m][n] + g_lds[1][1][m][n] + g_lds[2][1][m][n] + g_lds[3][1][m][n] + xp[1 * NH];
        float gg = g_lds[0][2][m][n] + g_lds[1][2][m][n] + g_lds[2][2][m][n] + g_lds[3][2][m][n] + xp[2 * NH];
        float go = g_lds[0][3][m][n] + g_lds[1][3][m][n] + g_lds[2][3][m][n] + g_lds[3][3][m][n] + xp[3 * NH];
        int idx = m * NH + n_base + n;
        float cn = sig_(gf) * c_prev[idx] + sig_(gi) * tanhf(gg);
        c_next[idx] = cn;
        h_next[idx] = sig_(go) * tanhf(cn);
    }
}

__global__ __launch_bounds__(512)
void lstm_step_l1(const float* __restrict__ x_in,   // layer-0 h (this step) [B x NH]
                  const float* __restrict__ Wih, const float* __restrict__ Whh,
                  const float* __restrict__ bih, const float* __restrict__ bhh,
                  const float* __restrict__ h_prev, const float* __restrict__ c_prev,
                  float* __restrict__ h_next, float* __restrict__ c_next,
                  float* __restrict__ hidden_t) {
    __shared__ float g_lds[4][4][BATCH][16];
    const int lane = threadIdx.x & 31;
    const int wave = threadIdx.x >> 5;
    const int gate = wave & 3;
    const int kq   = wave >> 2;
    const int n_base = blockIdx.x * 16;
    const int col_base = gate * NH + n_base;
    const int lm   = lane & 15;
    const int koff = (lane >> 4) * 2;
    const int kbeg = kq * KQ, kend = kbeg + KQ;

    v8f acc0 = {}, acc1 = {};
    wmma_pair_loop(x_in,   Wih + (size_t)(col_base + lm) * NH, kbeg, kend, acc0, acc1, lm, koff);
    wmma_pair_loop(h_prev, Whh + (size_t)(col_base + lm) * NH, kbeg, kend, acc0, acc1, lm, koff);

    const int row_half = (lane >> 4) * 8;
#pragma unroll
    for (int r = 0; r < 8; ++r) {
        g_lds[kq][gate][r + row_half][lm]      = acc0[r];
        g_lds[kq][gate][16 + r + row_half][lm] = acc1[r];
    }
    __syncthreads();
    {
        int e = threadIdx.x;
        int m = e >> 4, n = e & 15;
        int col = n_base + n;
        float gi = g_lds[0][0][m][n] + g_lds[1][0][m][n] + g_lds[2][0][m][n] + g_lds[3][0][m][n] + bih[0 * NH + col] + bhh[0 * NH + col];
        float gf = g_lds[0][1][m][n] + g_lds[1][1][m][n] + g_lds[2][1][m][n] + g_lds[3][1][m][n] + bih[1 * NH + col] + bhh[1 * NH + col];
        float gg = g_lds[0][2][m][n] + g_lds[1][2][m][n] + g_lds[2][2][m][n] + g_lds[3][2][m][n] + bih[2 * NH + col] + bhh[2 * NH + col];
        float go = g_lds[0][3][m][n] + g_lds[1][3][m][n] + g_lds[2][3][m][n] + g_lds[3][3][m][n] + bih[3 * NH + col] + bhh[3 * NH + col];
        int idx = m * NH + col;
        float cn = sig_(gf) * c_prev[idx] + sig_(gi) * tanhf(gg);
        float hn = sig_(go) * tanhf(cn);
        c_next[idx] = cn;
        h_next[idx] = hn;
        hidden_t[idx] = hn;
    }
}

// ---------------------------------------------------------------------------
// In-place log_softmax per row (one online max/sum pass + one write pass).
// ---------------------------------------------------------------------------
__global__ __launch_bounds__(256)
void log_softmax_inplace(float* __restrict__ logits, int N) {
    __shared__ float smax[256];
    __shared__ float ssum[256];
    float* row = logits + (size_t)blockIdx.x * N;
    float m = -INFINITY, s = 0.0f;
    for (int i = threadIdx.x; i < N; i += blockDim.x) {
        float x = row[i];
        if (x > m) { s *= __expf(m - x); m = x; }
        s += __expf(x - m);
    }
    smax[threadIdx.x] = m;
    ssum[threadIdx.x] = s;
    __syncthreads();
    for (int off = 128; off > 0; off >>= 1) {
        if (threadIdx.x < off) {
            float m1 = smax[threadIdx.x], s1 = ssum[threadIdx.x];
            float m2 = smax[threadIdx.x + off], s2 = ssum[threadIdx.x + off];
            float M = fmaxf(m1, m2);
            smax[threadIdx.x] = M;
            ssum[threadIdx.x] = s1 * __expf(m1 - M) + s2 * __expf(m2 - M);
        }
        __syncthreads();
    }
    const float lse = smax[0] + __logf(ssum[0]);
    for (int i = threadIdx.x; i < N; i += blockDim.x)
        row[i] = row[i] - lse;
}

__global__ void copy_f32(const float* __restrict__ src, float* __restrict__ dst, int n) {
    int i = blockIdx.x * blockDim.x + threadIdx.x;
    if (i < n) dst[i] = src[i];
}

// ---------------------------------------------------------------------------
extern "C" void kernel_launch(void* const* d_in, const int* in_sizes, int n_in,
                              void* d_out, int out_size, void* d_ws, size_t ws_size,
                              hipStream_t stream) {
    const float* x    = (const float*)d_in[0];   // [SEQ,B,NIN]
    const float* h0   = (const float*)d_in[1];   // [2,B,NH]
    const float* c0   = (const float*)d_in[2];
    const float* Wih0 = (const float*)d_in[3];   // [4H,NIN]
    const float* Whh0 = (const float*)d_in[4];   // [4H,NH]
    const float* bih0 = (const float*)d_in[5];
    const float* bhh0 = (const float*)d_in[6];
    const float* Wih1 = (const float*)d_in[7];
    const float* Whh1 = (const float*)d_in[8];
    const float* bih1 = (const float*)d_in[9];
    const float* bhh1 = (const float*)d_in[10];
    const float* Wl   = (const float*)d_in[11];  // [NOUT,NH]
    const float* bl   = (const float*)d_in[12];
    float* out = (float*)d_out;

    // Workspace layout
    float* ws = (float*)d_ws;
    size_t off = 0;
    float* X0pre  = ws + off; off += (size_t)MROWS * GDIM;  // [SEQ*B, 4H]
    float* hidden = ws + off; off += (size_t)MROWS * NH;    // [SEQ*B, NH]
    const size_t ST = (size_t)BATCH * NH;
    float* h0b[2] = { ws + off, ws + off + ST }; off += 2 * ST;
    float* c0b[2] = { ws + off, ws + off + ST }; off += 2 * ST;
    float* h1b[2] = { ws + off, ws + off + ST }; off += 2 * ST;
    float* c1b[2] = { ws + off, ws + off + ST }; off += 2 * ST;

    const int CP_BLK = 256;
    const int CP_GRID = (int)((ST + CP_BLK - 1) / CP_BLK);

    // Initial states (layer-major in the inputs)
    copy_f32<<<CP_GRID, CP_BLK, 0, stream>>>(h0,      h0b[0], (int)ST);
    copy_f32<<<CP_GRID, CP_BLK, 0, stream>>>(h0 + ST, h1b[0], (int)ST);
    copy_f32<<<CP_GRID, CP_BLK, 0, stream>>>(c0,      c0b[0], (int)ST);
    copy_f32<<<CP_GRID, CP_BLK, 0, stream>>>(c0 + ST, c1b[0], (int)ST);

    // X0pre = x @ Wih0^T + bih0 + bhh0  (time-independent, one big GEMM)
    {
        dim3 grid(GDIM / 64, MROWS / 256);
        gemm_nt_wmma<<<grid, 256, 0, stream>>>(x, Wih0, bih0, bhh0, X0pre,
                                               MROWS, GDIM, NIN);
    }

    // Sequential recurrence (state ping-pong avoids cross-WG hazards)
    for (int t = 0; t < SEQ; ++t) {
        const int cur = t & 1, nxt = cur ^ 1;
        lstm_step_l0<<<NH / 16, 512, 0, stream>>>(
            X0pre + (size_t)t * BATCH * GDIM, Whh0,
            h0b[cur], c0b[cur], h0b[nxt], c0b[nxt]);
        lstm_step_l1<<<NH / 16, 512, 0, stream>>>(
            h0b[nxt], Wih1, Whh1, bih1, bhh1,
            h1b[cur], c1b[cur], h1b[nxt], c1b[nxt],
            hidden + (size_t)t * BATCH * NH);
    }

    // Output head: logits -> logp region of d_out, then in-place log_softmax
    {
        dim3 grid(NOUT / 64, MROWS / 256);
        gemm_nt_wmma<<<grid, 256, 0, stream>>>(hidden, Wl, bl, nullptr, out,
                                               MROWS, NOUT, NH);
        log_softmax_inplace<<<MROWS, 256, 0, stream>>>(out, NOUT);
    }

    // Final states: after t=SEQ-1 the latest state sits in buffer index 0.
    const size_t HEAD = (size_t)MROWS * NOUT;
    copy_f32<<<CP_GRID, CP_BLK, 0, stream>>>(h0b[0], out + HEAD,          (int)ST);
    copy_f32<<<CP_GRID, CP_BLK, 0, stream>>>(h1b[0], out + HEAD + ST,     (int)ST);
    copy_f32<<<CP_GRID, CP_BLK, 0, stream>>>(c0b[0], out + HEAD + 2 * ST, (int)ST);
    copy_f32<<<CP_GRID, CP_BLK, 0, stream>>>(c1b[0], out + HEAD + 3 * ST, (int)ST);
    (void)in_sizes; (void)n_in; (void)out_size; (void)ws_size;
}